// LIF_63067299774994
// MI455X (gfx1250) — compile-verified
//
#include <hip/hip_runtime.h>
#include <stdint.h>
#include <math.h>

// LIF adaptive-threshold scan, B=16384 rows x L=2048 steps, fp32.
// Memory-bound floor: 128 MB in + 256 MB out = 384 MB @ 23.3 TB/s ~= 16.5 us.
// Sequential in L, parallel only in B -> one thread per row (512 waves).
// CDNA5 paths: async global->LDS input staging and async LDS->global output
// draining (both on ASYNCcnt, s_wait_asynccnt ordering), double-buffered.

constexpr int L_LEN = 2048;
constexpr int ROWS  = 64;            // rows per block == blockDim.x (2 waves)
constexpr int TCH   = 32;            // timesteps per LDS tile
constexpr int NT    = L_LEN / TCH;   // 64 tiles
constexpr int IN_STRIDE = 34;        // 32 cols + 2 pad: 8B-aligned rows, conflict-free reads
constexpr int ST_STRIDE = 36;        // 32 cols + 4 pad: 8B/16B-aligned rows

// 8 bytes per lane, GLOBAL -> LDS, GVS addressing (SADDR base + VADDR u32 offset).
__device__ __forceinline__ void async_load_b64(unsigned lds_addr, unsigned g_off,
                                               const float* base) {
  asm volatile("global_load_async_to_lds_b64 %0, %1, %2"
               :
               : "v"(lds_addr), "v"(g_off), "s"(base)
               : "memory");
}

// 8 bytes per lane, LDS -> GLOBAL.
__device__ __forceinline__ void async_store_b64(unsigned g_off, unsigned lds_addr,
                                                float* base) {
  asm volatile("global_store_async_from_lds_b64 %0, %1, %2"
               :
               : "v"(g_off), "v"(lds_addr), "s"(base)
               : "memory");
}

__device__ __forceinline__ void wait_async_16() {
#if __has_builtin(__builtin_amdgcn_s_wait_asynccnt)
  __builtin_amdgcn_s_wait_asynccnt(16);
#else
  asm volatile("s_wait_asynccnt 16" ::: "memory");
#endif
}

__device__ __forceinline__ void wait_async_all() {
#if __has_builtin(__builtin_amdgcn_s_wait_asynccnt)
  __builtin_amdgcn_s_wait_asynccnt(0);
#else
  asm volatile("s_wait_asynccnt 0" ::: "memory");
#endif
}

__global__ __launch_bounds__(ROWS) void lif_scan_kernel(const float* __restrict__ I,
                                                        float* __restrict__ out_spk,
                                                        float* __restrict__ out_ser,
                                                        int B) {
  __shared__ float in_buf[2][ROWS * IN_STRIDE];   // double-buffered input tiles
  __shared__ float stg_spk[2][ROWS * ST_STRIDE];  // double-buffered output staging
  __shared__ float stg_ser[2][ROWS * ST_STRIDE];

  const int tid  = threadIdx.x;
  const int row0 = blockIdx.x * ROWS;

  // Low 32 bits of a __shared__ generic address = wave-relative LDS byte offset.
  const unsigned lds_in[2]  = {(unsigned)(uintptr_t)&in_buf[0][0],
                               (unsigned)(uintptr_t)&in_buf[1][0]};
  const unsigned lds_spk[2] = {(unsigned)(uintptr_t)&stg_spk[0][0],
                               (unsigned)(uintptr_t)&stg_spk[1][0]};
  const unsigned lds_ser[2] = {(unsigned)(uintptr_t)&stg_ser[0][0],
                               (unsigned)(uintptr_t)&stg_ser[1][0]};

  // Tile = 64 rows x 32 floats = 1024 float2 -> 16 async b64 per wave.
  auto issue_tile_loads = [&](int t0, unsigned lbase) {
#pragma unroll
    for (int j = 0; j < 16; ++j) {
      int flat = j * ROWS + tid;          // float2 slot 0..1023
      int r    = flat >> 4;               // lanes 0-15 row r, 16-31 row r+1 (coalesced)
      int c2   = flat & 15;
      int rg   = row0 + r; if (rg > B - 1) rg = B - 1;
      async_load_b64(lbase + (unsigned)((r * IN_STRIDE + c2 * 2) * 4),
                     (unsigned)((rg * L_LEN + t0 + c2 * 2) * 4), I);
    }
  };

  // 32 async b64 stores per wave per tile (16 per output array).
  auto issue_tile_stores = [&](int t0, unsigned sbase, unsigned nbase) {
#pragma unroll
    for (int j = 0; j < 16; ++j) {
      int flat = j * ROWS + tid;
      int r    = flat >> 4;
      int c2   = flat & 15;
      int rg   = row0 + r;
      if (rg < B) {
        unsigned loff = (unsigned)((r * ST_STRIDE + c2 * 2) * 4);
        unsigned goff = (unsigned)((rg * L_LEN + t0 + c2 * 2) * 4);
        async_store_b64(goff, sbase + loff, out_spk);
        async_store_b64(goff, nbase + loff, out_ser);
      }
    }
  };

  // ---- prologue: tile 0 in flight ----
  issue_tile_loads(0, lds_in[0]);

  float v = 0.0f, a = 0.0f, snum = 0.0f;

  for (int t = 0; t < NT; ++t) {
    const int t0 = t * TCH;

    // Prefetch tile t+1, then bound the counter.
    // asynccnt<=16 proves (in-order within class): loads(t) resident AND
    // stores(t-2) drained (else the 32 newer stores(t-1) would all be pending).
    if (t + 1 < NT) {
      issue_tile_loads(t0 + TCH, lds_in[(t + 1) & 1]);
      wait_async_16();
    } else {
      wait_async_all();
    }
    __syncthreads();  // extend per-wave guarantees block-wide

    // ---- sequential LIF recurrence over this tile's 32 timesteps ----
    const float* in = &in_buf[t & 1][tid * IN_STRIDE];
    float* sp = &stg_spk[t & 1][tid * ST_STRIDE];
    float* se = &stg_ser[t & 1][tid * ST_STRIDE];

#pragma unroll
    for (int k = 0; k < TCH; ++k) {
#pragma clang fp contract(off)
      const float It    = in[k];
      const float th    = 1.0f + 1.5f * a;      // TH + BETA_ADAPT * a (old a)
      // XLA algebraic-simplifies v/20 -> v * fl(1/20); mirror that exactly.
      v = (v - v * 0.05f) + It;
      const float ssoft = 1.0f / (1.0f + expf(-10.0f * (v - th)));  // logistic expander form
      const float shard = (v >= th) ? 1.0f : 0.0f;
      const float s     = (shard - ssoft) + ssoft;  // straight-through forward value
      snum = snum + s;
      v = v * (1.0f - s) + (-0.5f) * s;             // detached reset, V_RESET=-0.5
      a = (a - a * 0.01f) + s;                      // a/100 -> a * fl(1/100)
      sp[k] = s;
      se[k] = snum;
    }
    __syncthreads();  // all waves' staging writes visible before async drain

    // ---- async drain: LDS staging -> global, overlapped with next compute ----
    issue_tile_stores(t0, lds_spk[t & 1], lds_ser[t & 1]);
  }

  wait_async_all();  // drain final tile's stores (S_ENDPGM also implies wait-idle)
}

extern "C" void kernel_launch(void* const* d_in, const int* in_sizes, int n_in,
                              void* d_out, int out_size, void* d_ws, size_t ws_size,
                              hipStream_t stream) {
  (void)n_in; (void)out_size; (void)d_ws; (void)ws_size;
  const float* I = (const float*)d_in[0];
  const int total = in_sizes[0];        // B * L
  const int B = total / L_LEN;
  float* out_spk = (float*)d_out;                       // spikes, (B, L) row-major
  float* out_ser = out_spk + (size_t)B * (size_t)L_LEN; // series, (B, L) row-major
  const int blocks = (B + ROWS - 1) / ROWS;
  lif_scan_kernel<<<blocks, ROWS, 0, stream>>>(I, out_spk, out_ser, B);
}